// AVStarTransformerEncoder_59133109731900
// MI455X (gfx1250) — compile-verified
//
#include <hip/hip_runtime.h>
#include <hip/hip_bf16.h>
#include <math.h>

typedef float v2f __attribute__((ext_vector_type(2)));
typedef float v8f __attribute__((ext_vector_type(8)));

#define DIM   256
#define FFN_D 2048
#define NH    8
#define NLV   3
#define NPTS  4
#define BB    2
#define LA_N  5
#define LV_N  8400

__device__ __forceinline__ float gelu_exact(float v)
{
    return 0.5f * v * (1.0f + erff(v * 0.70710678118654752f));
}

// ---------------------------------------------------------------------------
// fp32 GEMM via V_WMMA_F32_16X16X4_F32:  C = act(A @ W^T + bias)
// A: [M,K] row-major (optionally A+A2 elementwise), W: [N,K] row-major.
// One wave computes a 16x32 tile (two 16x16 WMMA tiles sharing the A frag).
// OOB rows/cols are clamped on load (they only feed unstored C elements)
// so the inner loop is branch-free: pure global_load_b64 + v_wmma.
// ---------------------------------------------------------------------------
template<bool HASA2, bool GELU>
__global__ __launch_bounds__(128)
void gemm_f32_wmma(const float* __restrict__ A, const float* __restrict__ A2,
                   const float* __restrict__ W, const float* __restrict__ bias,
                   float* __restrict__ C, int M, int N, int K)
{
    const int lane = threadIdx.x & 31;
    const int wave = threadIdx.y;
    const int tilesN2 = (N + 31) >> 5;            // pairs of 16-col tiles
    const int tilesM  = (M + 15) >> 4;
    int t = blockIdx.x * 4 + wave;
    if (t >= tilesN2 * tilesM) return;            // wave-uniform exit
    const int tn = t % tilesN2;
    const int tm = t / tilesN2;

    const int half = lane >> 4;                   // 0: K+0/1, 1: K+2/3
    const int l16  = lane & 15;

    const int arow  = tm * 16 + l16;
    const int bcol0 = tn * 32 + l16;
    const int bcol1 = bcol0 + 16;
    const int arc = (arow  < M) ? arow  : (M - 1);
    const int bc0 = (bcol0 < N) ? bcol0 : (N - 1);
    const int bc1 = (bcol1 < N) ? bcol1 : (N - 1);

    const float* __restrict__ Ap  = A + (size_t)arc * K + 2 * half;
    const float* __restrict__ A2p = HASA2 ? (A2 + (size_t)arc * K + 2 * half) : nullptr;
    const float* __restrict__ Wp0 = W + (size_t)bc0 * K + 2 * half;
    const float* __restrict__ Wp1 = W + (size_t)bc1 * K + 2 * half;

    v8f acc0 = {}, acc1 = {};
    for (int k = 0; k < K; k += 8) {
        v2f a0 = *(const v2f*)(Ap + k);
        v2f a1 = *(const v2f*)(Ap + k + 4);
        if (HASA2) {
            a0 += *(const v2f*)(A2p + k);
            a1 += *(const v2f*)(A2p + k + 4);
        }
        v2f b00 = *(const v2f*)(Wp0 + k);
        v2f b01 = *(const v2f*)(Wp0 + k + 4);
        v2f b10 = *(const v2f*)(Wp1 + k);
        v2f b11 = *(const v2f*)(Wp1 + k + 4);
        acc0 = __builtin_amdgcn_wmma_f32_16x16x4_f32(false, a0, false, b00, (short)0, acc0, false, false);
        acc1 = __builtin_amdgcn_wmma_f32_16x16x4_f32(false, a0, false, b10, (short)0, acc1, false, false);
        acc0 = __builtin_amdgcn_wmma_f32_16x16x4_f32(false, a1, false, b01, (short)0, acc0, false, false);
        acc1 = __builtin_amdgcn_wmma_f32_16x16x4_f32(false, a1, false, b11, (short)0, acc1, false, false);
    }

    const float bv0 = bias ? bias[bc0] : 0.0f;
    const float bv1 = bias ? bias[bc1] : 0.0f;
#pragma unroll
    for (int r = 0; r < 8; ++r) {
        const int crow = tm * 16 + r + 8 * half;
        if (crow < M) {
            const size_t base = (size_t)crow * N;
            if (bcol0 < N) {
                float v = acc0[r] + bv0;
                if (GELU) v = gelu_exact(v);
                C[base + bcol0] = v;
            }
            if (bcol1 < N) {
                float v = acc1[r] + bv1;
                if (GELU) v = gelu_exact(v);
                C[base + bcol1] = v;
            }
        }
    }
}

// ---------------------------------------------------------------------------
// Reference points: one thread per (b,q); writes ref[b,q,l,2] for all levels.
// ---------------------------------------------------------------------------
__global__ void ref_points_kernel(const int* __restrict__ shapes,
                                  const int* __restrict__ lsi,
                                  const float* __restrict__ vr,
                                  float* __restrict__ ref)
{
    int i = blockIdx.x * blockDim.x + threadIdx.x;
    if (i >= BB * LV_N) return;
    int b = i / LV_N, q = i % LV_N;
    int lvl = 2;
    if (q < lsi[1]) lvl = 0; else if (q < lsi[2]) lvl = 1;
    int H = shapes[lvl * 2], W = shapes[lvl * 2 + 1];
    int o = q - lsi[lvl];
    int iy = o / W, ix = o % W;
    float ry = (iy + 0.5f) / (vr[(b * NLV + lvl) * 2 + 1] * (float)H);
    float rx = (ix + 0.5f) / (vr[(b * NLV + lvl) * 2 + 0] * (float)W);
#pragma unroll
    for (int l = 0; l < NLV; ++l) {
        ref[(((size_t)(b * LV_N + q)) * NLV + l) * 2 + 0] = rx * vr[(b * NLV + l) * 2 + 0];
        ref[(((size_t)(b * LV_N + q)) * NLV + l) * 2 + 1] = ry * vr[(b * NLV + l) * 2 + 1];
    }
}

// ---------------------------------------------------------------------------
// MS-deformable sampling: one wave per (b,q,h), lane = head-channel (0..31).
// ---------------------------------------------------------------------------
__global__ __launch_bounds__(256)
void msdeform_sample_kernel(const float* __restrict__ value,
                            const float* __restrict__ off,
                            const float* __restrict__ aw,
                            const float* __restrict__ ref,
                            const int* __restrict__ shapes,
                            const int* __restrict__ lsi,
                            float* __restrict__ out)
{
    int wid  = (blockIdx.x * blockDim.x + threadIdx.x) >> 5;
    int lane = threadIdx.x & 31;
    if (wid >= BB * LV_N * NH) return;
    int h = wid % NH;
    int q = (wid / NH) % LV_N;
    int b = wid / (NH * LV_N);

    // softmax over NL*NPTS = 12 attention logits
    const float* awp = aw + ((size_t)(b * LV_N + q) * NH + h) * (NLV * NPTS);
    float lg[NLV * NPTS], mx = -1e30f;
#pragma unroll
    for (int j = 0; j < NLV * NPTS; ++j) { lg[j] = awp[j]; mx = fmaxf(mx, lg[j]); }
    float s = 0.0f;
#pragma unroll
    for (int j = 0; j < NLV * NPTS; ++j) { lg[j] = expf(lg[j] - mx); s += lg[j]; }
    const float inv = 1.0f / s;

    const float* offp = off + ((size_t)(b * LV_N + q) * NH + h) * (NLV * NPTS * 2);
    const float* refp = ref + ((size_t)(b * LV_N + q)) * (NLV * 2);

    float accv = 0.0f;
#pragma unroll
    for (int l = 0; l < NLV; ++l) {
        int H = shapes[l * 2], W = shapes[l * 2 + 1];
        int start = lsi[l];
        float rx = refp[l * 2 + 0], ry = refp[l * 2 + 1];
        const float* vbase = value + ((size_t)b * LV_N + start) * DIM + h * 32 + lane;
#pragma unroll
        for (int p = 0; p < NPTS; ++p) {
            float ox = offp[(l * NPTS + p) * 2 + 0];
            float oy = offp[(l * NPTS + p) * 2 + 1];
            float lx = rx + ox / (float)W;
            float ly = ry + oy / (float)H;
            float x = lx * (float)W - 0.5f;
            float y = ly * (float)H - 0.5f;
            float x0f = floorf(x), y0f = floorf(y);
            float fx = x - x0f, fy = y - y0f;
            int x0 = (int)x0f, y0 = (int)y0f;
            float sample = 0.0f;
#pragma unroll
            for (int c = 0; c < 4; ++c) {
                int xi = x0 + (c & 1);
                int yi = y0 + (c >> 1);
                float wc = ((c & 1) ? fx : 1.0f - fx) * ((c >> 1) ? fy : 1.0f - fy);
                bool valid = (xi >= 0) && (xi < W) && (yi >= 0) && (yi < H);
                int xc = min(max(xi, 0), W - 1);
                int yc = min(max(yi, 0), H - 1);
                float v = vbase[(size_t)(yc * W + xc) * DIM];
                sample += valid ? (wc * v) : 0.0f;
            }
            accv += (lg[l * NPTS + p] * inv) * sample;
        }
    }
    out[((size_t)(b * LV_N + q)) * DIM + h * 32 + lane] = accv;
}

__device__ __forceinline__ float wave_sum(float v)
{
#pragma unroll
    for (int o = 16; o > 0; o >>= 1) v += __shfl_xor(v, o, 32);
    return v;
}

// ---------------------------------------------------------------------------
// Cross-attention (8 heads, Lk = LA_N): one wave per (b,q,h), lane = channel.
// ---------------------------------------------------------------------------
__global__ __launch_bounds__(256)
void cross_attn_kernel(const float* __restrict__ q, const float* __restrict__ k,
                       const float* __restrict__ v, float* __restrict__ out)
{
    int wid  = (blockIdx.x * blockDim.x + threadIdx.x) >> 5;
    int lane = threadIdx.x & 31;
    if (wid >= BB * LV_N * NH) return;
    int h  = wid % NH;
    int qq = (wid / NH) % LV_N;
    int b  = wid / (NH * LV_N);

    float qv = q[((size_t)(b * LV_N + qq)) * DIM + h * 32 + lane];
    float sc[LA_N], mx = -1e30f;
#pragma unroll
    for (int j = 0; j < LA_N; ++j) {
        float kv = k[((size_t)(b * LA_N + j)) * DIM + h * 32 + lane];
        float d = wave_sum(qv * kv) * 0.17677669529663687f;   // 1/sqrt(32)
        sc[j] = d; mx = fmaxf(mx, d);
    }
    float s = 0.0f;
#pragma unroll
    for (int j = 0; j < LA_N; ++j) { sc[j] = expf(sc[j] - mx); s += sc[j]; }
    float inv = 1.0f / s;
    float accv = 0.0f;
#pragma unroll
    for (int j = 0; j < LA_N; ++j)
        accv += sc[j] * inv * v[((size_t)(b * LA_N + j)) * DIM + h * 32 + lane];
    out[((size_t)(b * LV_N + qq)) * DIM + h * 32 + lane] = accv;
}

// ---------------------------------------------------------------------------
// src_a self-attention (1 head, E=256): one wave per (b,q).
// ---------------------------------------------------------------------------
__global__ __launch_bounds__(64)
void a_self_attn_kernel(const float* __restrict__ q, const float* __restrict__ k,
                        const float* __restrict__ v, float* __restrict__ out)
{
    int wid  = (blockIdx.x * blockDim.x + threadIdx.x) >> 5;
    int lane = threadIdx.x & 31;
    if (wid >= BB * LA_N) return;
    int qq = wid % LA_N, b = wid / LA_N;
    const float* qp = q + ((size_t)(b * LA_N + qq)) * DIM;

    float sc[LA_N], mx = -1e30f;
#pragma unroll
    for (int j = 0; j < LA_N; ++j) {
        const float* kp = k + ((size_t)(b * LA_N + j)) * DIM;
        float part = 0.0f;
#pragma unroll
        for (int c = 0; c < DIM / 32; ++c) part += qp[lane + 32 * c] * kp[lane + 32 * c];
        float d = wave_sum(part) * 0.0625f;                   // 1/sqrt(256)
        sc[j] = d; mx = fmaxf(mx, d);
    }
    float s = 0.0f;
#pragma unroll
    for (int j = 0; j < LA_N; ++j) { sc[j] = expf(sc[j] - mx); s += sc[j]; }
    float inv = 1.0f / s;
#pragma unroll
    for (int c = 0; c < DIM / 32; ++c) {
        float accv = 0.0f;
#pragma unroll
        for (int j = 0; j < LA_N; ++j)
            accv += sc[j] * inv * v[((size_t)(b * LA_N + j)) * DIM + lane + 32 * c];
        out[((size_t)(b * LA_N + qq)) * DIM + lane + 32 * c] = accv;
    }
}

// ---------------------------------------------------------------------------
// y = LayerNorm(x + delta) * g + b, in place on x. One block (256 thr) / row.
// ---------------------------------------------------------------------------
__global__ __launch_bounds__(256)
void ln_residual_kernel(float* __restrict__ x, const float* __restrict__ delta,
                        const float* __restrict__ g, const float* __restrict__ bta,
                        int rows)
{
    __shared__ float red[256];
    int row = blockIdx.x;
    if (row >= rows) return;
    int t = threadIdx.x;
    size_t base = (size_t)row * DIM;
    float v = x[base + t] + delta[base + t];
    red[t] = v; __syncthreads();
#pragma unroll
    for (int o = 128; o > 0; o >>= 1) { if (t < o) red[t] += red[t + o]; __syncthreads(); }
    float mu = red[0] * (1.0f / DIM); __syncthreads();
    float d = v - mu;
    red[t] = d * d; __syncthreads();
#pragma unroll
    for (int o = 128; o > 0; o >>= 1) { if (t < o) red[t] += red[t + o]; __syncthreads(); }
    float var = red[0] * (1.0f / DIM);
    float r = rsqrtf(var + 1e-5f);
    x[base + t] = d * r * g[t] + bta[t];
}

// ---------------------------------------------------------------------------
extern "C" void kernel_launch(void* const* d_in, const int* in_sizes, int n_in,
                              void* d_out, int out_size, void* d_ws, size_t ws_size,
                              hipStream_t stream)
{
    const float* src_a = (const float*)d_in[0];
    const float* src_v = (const float*)d_in[1];
    const int*   shapes = (const int*)d_in[2];
    const int*   lsi    = (const int*)d_in[3];
    const float* vr     = (const float*)d_in[4];
    const float* pos    = (const float*)d_in[5];

    const int MV = BB * LV_N;   // 16800
    const int MA = BB * LA_N;   // 10

    float* outA = (float*)d_out;
    float* outV = outA + (size_t)MA * DIM;
    float* refO = outV + (size_t)MV * DIM;

    float* ws = (float*)d_ws;
    size_t o = 0;
    float* bufQ   = ws + o; o += (size_t)MV * DIM;     // 4.30M floats
    float* bufV   = ws + o; o += (size_t)MV * DIM;     // 4.30M
    float* bufT   = ws + o; o += (size_t)MV * DIM;     // 4.30M (also FFN hidden chunks)
    float* bufOff = ws + o; o += (size_t)MV * 192;
    float* bufAw  = ws + o; o += (size_t)MV * 96;
    float* aQ = ws + o; o += (size_t)MA * DIM;
    float* aK = ws + o; o += (size_t)MA * DIM;
    float* aV = ws + o; o += (size_t)MA * DIM;
    float* aT = ws + o; o += (size_t)MA * DIM;
    float* aH = ws + o; o += (size_t)MA * FFN_D;

    auto gemm = [&](const float* A, const float* A2, const float* W, const float* bias,
                    float* C, int M, int N, int K, int act) {
        int tiles = ((M + 15) / 16) * ((N + 31) / 32);
        dim3 grid((tiles + 3) / 4), blk(32, 4);
        if (A2)           gemm_f32_wmma<true,  false><<<grid, blk, 0, stream>>>(A, A2, W, bias, C, M, N, K);
        else if (act == 1) gemm_f32_wmma<false, true ><<<grid, blk, 0, stream>>>(A, nullptr, W, bias, C, M, N, K);
        else              gemm_f32_wmma<false, false><<<grid, blk, 0, stream>>>(A, nullptr, W, bias, C, M, N, K);
    };
    auto P = [&](int layer, int idx) -> const float* {
        return (const float*)d_in[6 + layer * 34 + idx];
    };

    // running state lives in d_out
    hipMemcpyAsync(outA, src_a, (size_t)MA * DIM * sizeof(float), hipMemcpyDeviceToDevice, stream);
    hipMemcpyAsync(outV, src_v, (size_t)MV * DIM * sizeof(float), hipMemcpyDeviceToDevice, stream);

    ref_points_kernel<<<dim3((BB * LV_N + 255) / 256), dim3(256), 0, stream>>>(shapes, lsi, vr, refO);

    const int wavesVH = BB * LV_N * NH;
    const int blkVH   = (wavesVH * 32 + 255) / 256;

    for (int L = 0; L < 2; ++L) {
        const float *off_w = P(L,0), *off_b = P(L,1), *aw_w = P(L,2), *aw_b = P(L,3),
                    *val_w = P(L,4), *val_b = P(L,5), *out_w = P(L,6), *out_b = P(L,7),
                    *nvg = P(L,8), *nvb = P(L,9),
                    *a_in_w = P(L,10), *a_in_b = P(L,11), *a_out_w = P(L,12), *a_out_b = P(L,13),
                    *va_in_w = P(L,14), *va_in_b = P(L,15), *va_out_w = P(L,16), *va_out_b = P(L,17),
                    *nag = P(L,18), *nab = P(L,19), *nvag = P(L,20), *nvab = P(L,21),
                    *v1_w = P(L,22), *v1_b = P(L,23), *v2_w = P(L,24), *v2_b = P(L,25),
                    *nvfg = P(L,26), *nvfb = P(L,27),
                    *a1_w = P(L,28), *a1_b = P(L,29), *a2_w = P(L,30), *a2_b = P(L,31),
                    *nafg = P(L,32), *nafb = P(L,33);

        // ---- MS-deformable attention on src_v ----
        gemm(outV, nullptr, val_w, val_b, bufV,   MV, DIM, DIM, 0);  // value proj (no pos)
        gemm(outV, pos,     off_w, off_b, bufOff, MV, 192, DIM, 0);  // offsets (query = v+pos)
        gemm(outV, pos,     aw_w,  aw_b,  bufAw,  MV,  96, DIM, 0);  // attn weights
        msdeform_sample_kernel<<<dim3(blkVH), dim3(256), 0, stream>>>(
            bufV, bufOff, bufAw, refO, shapes, lsi, bufT);
        gemm(bufT, nullptr, out_w, out_b, bufQ, MV, DIM, DIM, 0);
        ln_residual_kernel<<<dim3(MV), dim3(256), 0, stream>>>(outV, bufQ, nvg, nvb, MV);

        // ---- src_a self-attention (1 head) ----
        gemm(outA, nullptr, a_in_w,             a_in_b,       aQ, MA, DIM, DIM, 0);
        gemm(outA, nullptr, a_in_w + 256 * 256, a_in_b + 256, aK, MA, DIM, DIM, 0);
        gemm(outA, nullptr, a_in_w + 512 * 256, a_in_b + 512, aV, MA, DIM, DIM, 0);
        a_self_attn_kernel<<<dim3((BB * LA_N * 32 + 63) / 64), dim3(64), 0, stream>>>(aQ, aK, aV, aT);
        gemm(aT, nullptr, a_out_w, a_out_b, aQ, MA, DIM, DIM, 0);
        ln_residual_kernel<<<dim3(MA), dim3(256), 0, stream>>>(outA, aQ, nag, nab, MA);

        // ---- cross attention: src_v queries attend to src_a (8 heads) ----
        gemm(outV, nullptr, va_in_w,             va_in_b,       bufQ, MV, DIM, DIM, 0);
        gemm(outA, nullptr, va_in_w + 256 * 256, va_in_b + 256, aK,   MA, DIM, DIM, 0);
        gemm(outA, nullptr, va_in_w + 512 * 256, va_in_b + 512, aV,   MA, DIM, DIM, 0);
        cross_attn_kernel<<<dim3(blkVH), dim3(256), 0, stream>>>(bufQ, aK, aV, bufT);
        gemm(bufT, nullptr, va_out_w, va_out_b, bufQ, MV, DIM, DIM, 0);
        ln_residual_kernel<<<dim3(MV), dim3(256), 0, stream>>>(outV, bufQ, nvag, nvab, MV);

        // ---- FFN on src_v (row-chunked; hidden chunk reuses bufT) ----
        const int CH = 2048;
        for (int r0 = 0; r0 < MV; r0 += CH) {
            int rc = (MV - r0 < CH) ? (MV - r0) : CH;
            gemm(outV + (size_t)r0 * DIM, nullptr, v1_w, v1_b, bufT, rc, FFN_D, DIM, 1);
            gemm(bufT, nullptr, v2_w, v2_b, bufQ + (size_t)r0 * DIM, rc, DIM, FFN_D, 0);
        }
        ln_residual_kernel<<<dim3(MV), dim3(256), 0, stream>>>(outV, bufQ, nvfg, nvfb, MV);

        // ---- FFN on src_a ----
        gemm(outA, nullptr, a1_w, a1_b, aH, MA, FFN_D, DIM, 1);
        gemm(aH,   nullptr, a2_w, a2_b, aQ, MA, DIM, FFN_D, 0);
        ln_residual_kernel<<<dim3(MA), dim3(256), 0, stream>>>(outA, aQ, nafg, nafb, MA);
    }
}